// Decoder_67087389163554
// MI455X (gfx1250) — compile-verified
//
#include <hip/hip_runtime.h>
#include <hip/hip_bf16.h>

// Problem constants (from reference): V=32000, H=E=1024, B=64, S=128
constexpr int Vn = 32000;
constexpr int Hn = 1024;
constexpr int En = 1024;
constexpr int Bn = 64;
constexpr int Sn = 128;
constexpr int XK = En + 2 * Hn;   // 3072: [emb | context | h0]
constexpr int GN = 4 * Hn;        // 4096 gate columns

typedef float v2f __attribute__((ext_vector_type(2)));
typedef float v8f __attribute__((ext_vector_type(8)));

__device__ __forceinline__ v8f wmma_f32(v2f a, v2f b, v8f c) {
  // V_WMMA_F32_16X16X4_F32: D(16x16,f32) = A(16x4) * B(4x16) + C
  return __builtin_amdgcn_wmma_f32_16x16x4_f32(
      /*neg_a=*/false, a, /*neg_b=*/false, b,
      /*c_mod=*/(short)0, c, /*reuse_a=*/false, /*reuse_b=*/false);
}

// Single-use weight streams: NT hint keeps the 192MB L2 free for reused data
// (encoder_outputs, h1/x_cat, logits, emb rows).
__device__ __forceinline__ v2f nt_load2(const float* p) {
  return __builtin_nontemporal_load((const v2f*)p);
}

__device__ __forceinline__ float wave_sum(float v) {
#pragma unroll
  for (int m = 16; m >= 1; m >>= 1) v += __shfl_xor(v, m, 32);
  return v;
}

__device__ __forceinline__ float sigmoidf_(float x) {
  return 1.0f / (1.0f + __expf(-x));
}

// ---------------------------------------------------------------------------
// Kernel 1: attention scores + softmax + context, plus x_cat assembly.
// One block per batch row, 256 threads (8 waves of 32).
// ---------------------------------------------------------------------------
__global__ void attn_ctx_kernel(const int* __restrict__ input_batch,
                                const float* __restrict__ prev_h,
                                const float* __restrict__ enc,
                                const float* __restrict__ emb,
                                const float* __restrict__ attn_w,
                                const float* __restrict__ attn_b,
                                float* __restrict__ x_cat,
                                float* __restrict__ attn_out) {
  const int b = blockIdx.x;
  const int tid = threadIdx.x;
  const int lane = tid & 31;
  const int wave = tid >> 5;

  __shared__ float s_scores[Sn];
  __shared__ float s_red[256];
  __shared__ float s_hpart;

  const float* encb = enc + (size_t)b * Sn * Hn;
  const float* h0b = prev_h + (size_t)b * Hn;

  // e_part[s] = dot(enc[b,s,:], attn_w[H:])  -- 8 waves x 16 scores
#pragma unroll 1
  for (int i = 0; i < Sn / 8; ++i) {
    const int s = wave * (Sn / 8) + i;
    const float* row = encb + (size_t)s * Hn;
    float acc = 0.f;
#pragma unroll 4
    for (int j = lane; j < Hn; j += 32) acc += row[j] * attn_w[Hn + j];
    acc = wave_sum(acc);
    if (lane == 0) s_scores[s] = acc;
  }

  // h_part = dot(h0[b], attn_w[:H]) -- block reduce
  float hp = 0.f;
  for (int j = tid; j < Hn; j += 256) hp += h0b[j] * attn_w[j];
  s_red[tid] = hp;
  __syncthreads();
  for (int off = 128; off >= 1; off >>= 1) {
    if (tid < off) s_red[tid] += s_red[tid + off];
    __syncthreads();
  }
  if (tid == 0) s_hpart = s_red[0] + attn_b[0];
  __syncthreads();

  // softmax over the 128 scores
  float sc = (tid < Sn) ? (s_scores[tid] + s_hpart) : -3.4e38f;
  s_red[tid] = sc;
  __syncthreads();
  for (int off = 128; off >= 1; off >>= 1) {
    if (tid < off) s_red[tid] = fmaxf(s_red[tid], s_red[tid + off]);
    __syncthreads();
  }
  const float mx = s_red[0];
  __syncthreads();
  float ex = (tid < Sn) ? __expf(sc - mx) : 0.f;
  s_red[tid] = ex;
  __syncthreads();
  for (int off = 128; off >= 1; off >>= 1) {
    if (tid < off) s_red[tid] += s_red[tid + off];
    __syncthreads();
  }
  const float inv = 1.0f / s_red[0];
  __syncthreads();
  if (tid < Sn) {
    const float w = ex * inv;
    s_scores[tid] = w;
    attn_out[(size_t)b * Sn + tid] = w;
  }
  __syncthreads();

  // context[h] = sum_s w[s] * enc[b,s,h]  (coalesced along h)
  float* xb = x_cat + (size_t)b * XK;
  for (int h = tid; h < Hn; h += 256) {
    float acc = 0.f;
#pragma unroll 8
    for (int s = 0; s < Sn; ++s) acc += s_scores[s] * encb[(size_t)s * Hn + h];
    xb[En + h] = acc;
  }

  // x_cat = [emb[token] | context | h0]
  const int tok = input_batch[b];
  const float* erow = emb + (size_t)tok * En;
  for (int j = tid; j < Hn; j += 256) {
    xb[j] = erow[j];
    xb[En + Hn + j] = h0b[j];
  }
}

// ---------------------------------------------------------------------------
// Kernel 2: gates = x_cat @ [W_ih|W_hh]^T + b_ih + b_hh   (64 x 4096)
// 32 blocks x 256 threads; each wave owns a 64x16 strip (4 stacked 16x16 C).
// ---------------------------------------------------------------------------
__global__ void gates_gemm_kernel(const float* __restrict__ x_cat,
                                  const float* __restrict__ W_ih,
                                  const float* __restrict__ W_hh,
                                  const float* __restrict__ b_ih,
                                  const float* __restrict__ b_hh,
                                  float* __restrict__ gates) {
  const int tid = threadIdx.x;
  const int lane = tid & 31;
  const int wave = tid >> 5;
  const int nl = (blockIdx.x * 8 + wave) * 16 + (lane & 15);  // my N column
  const int m = lane & 15;                                    // A row in tile
  const int kp = (lane >> 4) * 2;                             // K pair 0 or 2

  v8f c0 = {}, c1 = {}, c2 = {}, c3 = {};

  // Phase 1: K = 0..2047 against W_ih (row stride 2048), A cols [0,2048)
  {
    const float* Brow = W_ih + (size_t)nl * 2048;
#pragma unroll 4
    for (int kb = 0; kb < 2048; kb += 4) {
      const int k = kb + kp;
      v2f bf = nt_load2(Brow + k);                    // single-use weight stream
      v2f a0 = *(const v2f*)(x_cat + (size_t)(m + 0)  * XK + k);
      v2f a1 = *(const v2f*)(x_cat + (size_t)(m + 16) * XK + k);
      v2f a2 = *(const v2f*)(x_cat + (size_t)(m + 32) * XK + k);
      v2f a3 = *(const v2f*)(x_cat + (size_t)(m + 48) * XK + k);
      c0 = wmma_f32(a0, bf, c0);
      c1 = wmma_f32(a1, bf, c1);
      c2 = wmma_f32(a2, bf, c2);
      c3 = wmma_f32(a3, bf, c3);
    }
  }
  // Phase 2: K = 0..1023 against W_hh (row stride 1024), A cols [2048,3072)
  {
    const float* Brow = W_hh + (size_t)nl * 1024;
#pragma unroll 4
    for (int kb = 0; kb < 1024; kb += 4) {
      const int k = kb + kp;
      v2f bf = nt_load2(Brow + k);                    // single-use weight stream
      v2f a0 = *(const v2f*)(x_cat + (size_t)(m + 0)  * XK + 2048 + k);
      v2f a1 = *(const v2f*)(x_cat + (size_t)(m + 16) * XK + 2048 + k);
      v2f a2 = *(const v2f*)(x_cat + (size_t)(m + 32) * XK + 2048 + k);
      v2f a3 = *(const v2f*)(x_cat + (size_t)(m + 48) * XK + 2048 + k);
      c0 = wmma_f32(a0, bf, c0);
      c1 = wmma_f32(a1, bf, c1);
      c2 = wmma_f32(a2, bf, c2);
      c3 = wmma_f32(a3, bf, c3);
    }
  }

  const float bias = b_ih[nl] + b_hh[nl];
  const int rbase = (lane >> 4) * 8;  // lanes 16-31 hold rows +8
#pragma unroll
  for (int r = 0; r < 8; ++r) {
    const int row = r + rbase;
    gates[(size_t)(row + 0)  * GN + nl] = c0[r] + bias;
    gates[(size_t)(row + 16) * GN + nl] = c1[r] + bias;
    gates[(size_t)(row + 32) * GN + nl] = c2[r] + bias;
    gates[(size_t)(row + 48) * GN + nl] = c3[r] + bias;
  }
}

// ---------------------------------------------------------------------------
// Kernel 3: LSTM pointwise -> h1, c1 (with zero-length mask)
// ---------------------------------------------------------------------------
__global__ void lstm_kernel(const float* __restrict__ gates,
                            const float* __restrict__ prev_c,
                            const int* __restrict__ lengths,
                            float* __restrict__ h1,
                            float* __restrict__ c1) {
  const int b = blockIdx.x;
  const int tid = threadIdx.x;
  const float* g = gates + (size_t)b * GN;
  const bool z = (lengths[b] == 0);
  for (int h = tid; h < Hn; h += 256) {
    const float ig = sigmoidf_(g[h]);
    const float fg = sigmoidf_(g[Hn + h]);
    const float gg = tanhf(g[2 * Hn + h]);
    const float og = sigmoidf_(g[3 * Hn + h]);
    float c = fg * prev_c[(size_t)b * Hn + h] + ig * gg;
    float hh = og * tanhf(c);
    if (z) { c = 0.f; hh = 0.f; }
    h1[(size_t)b * Hn + h] = hh;
    c1[(size_t)b * Hn + h] = c;
  }
}

// ---------------------------------------------------------------------------
// Kernel 4: logits = h1 @ W_out^T + b_out   (64 x 32000)  -- streams 131 MB
// 250 blocks x 256 threads; wave = 64x16 strip, K=1024 f32 WMMA loop.
// ---------------------------------------------------------------------------
__global__ void logits_gemm_kernel(const float* __restrict__ h1,
                                   const float* __restrict__ W_out,
                                   const float* __restrict__ b_out,
                                   float* __restrict__ logits) {
  const int tid = threadIdx.x;
  const int lane = tid & 31;
  const int wave = tid >> 5;
  const int nl = (blockIdx.x * 8 + wave) * 16 + (lane & 15);
  const int m = lane & 15;
  const int kp = (lane >> 4) * 2;

  v8f c0 = {}, c1 = {}, c2 = {}, c3 = {};
  const float* Brow = W_out + (size_t)nl * Hn;

#pragma unroll 4
  for (int kb = 0; kb < Hn; kb += 4) {
    const int k = kb + kp;
    v2f bf = nt_load2(Brow + k);                             // NT: 131MB stream
    v2f a0 = *(const v2f*)(h1 + (size_t)(m + 0)  * Hn + k);  // L2/WGP$-resident
    v2f a1 = *(const v2f*)(h1 + (size_t)(m + 16) * Hn + k);
    v2f a2 = *(const v2f*)(h1 + (size_t)(m + 32) * Hn + k);
    v2f a3 = *(const v2f*)(h1 + (size_t)(m + 48) * Hn + k);
    c0 = wmma_f32(a0, bf, c0);
    c1 = wmma_f32(a1, bf, c1);
    c2 = wmma_f32(a2, bf, c2);
    c3 = wmma_f32(a3, bf, c3);
  }

  const float bias = b_out[nl];
  const int rbase = (lane >> 4) * 8;
#pragma unroll
  for (int r = 0; r < 8; ++r) {
    const int row = r + rbase;
    logits[(size_t)(row + 0)  * Vn + nl] = c0[r] + bias;
    logits[(size_t)(row + 16) * Vn + nl] = c1[r] + bias;
    logits[(size_t)(row + 32) * Vn + nl] = c2[r] + bias;
    logits[(size_t)(row + 48) * Vn + nl] = c3[r] + bias;
  }
}

// ---------------------------------------------------------------------------
// Kernel 5: row-wise log_softmax over V=32000 (one 512-thread block per row)
// ---------------------------------------------------------------------------
__global__ void logsoftmax_kernel(const float* __restrict__ logits,
                                  float* __restrict__ out) {
  const int b = blockIdx.x;
  const int tid = threadIdx.x;
  __shared__ float s_red[512];
  const float* row = logits + (size_t)b * Vn;

  float m = -3.4e38f;
  for (int j = tid; j < Vn; j += 512) m = fmaxf(m, row[j]);
  s_red[tid] = m;
  __syncthreads();
  for (int off = 256; off >= 1; off >>= 1) {
    if (tid < off) s_red[tid] = fmaxf(s_red[tid], s_red[tid + off]);
    __syncthreads();
  }
  m = s_red[0];
  __syncthreads();

  float sum = 0.f;
  for (int j = tid; j < Vn; j += 512) sum += __expf(row[j] - m);
  s_red[tid] = sum;
  __syncthreads();
  for (int off = 256; off >= 1; off >>= 1) {
    if (tid < off) s_red[tid] += s_red[tid + off];
    __syncthreads();
  }
  const float lse = m + __logf(s_red[0]);

  for (int j = tid; j < Vn; j += 512)
    out[(size_t)b * Vn + j] = row[j] - lse;
}

// ---------------------------------------------------------------------------
extern "C" void kernel_launch(void* const* d_in, const int* in_sizes, int n_in,
                              void* d_out, int out_size, void* d_ws, size_t ws_size,
                              hipStream_t stream) {
  (void)in_sizes; (void)n_in; (void)out_size; (void)ws_size;

  const int*   input_batch = (const int*)d_in[0];
  const float* prev_h      = (const float*)d_in[1];
  const float* prev_c      = (const float*)d_in[2];
  const float* enc         = (const float*)d_in[3];
  const int*   lengths     = (const int*)d_in[4];
  const float* emb         = (const float*)d_in[5];
  const float* attn_w      = (const float*)d_in[6];
  const float* attn_b      = (const float*)d_in[7];
  const float* W_ih        = (const float*)d_in[8];
  const float* W_hh        = (const float*)d_in[9];
  const float* b_ih        = (const float*)d_in[10];
  const float* b_hh        = (const float*)d_in[11];
  const float* W_out       = (const float*)d_in[12];
  const float* b_out       = (const float*)d_in[13];

  // Output layout: logprobs | h1 | c1 | attn_weights (flat, return order)
  float* out      = (float*)d_out;
  float* logprobs = out;
  float* h1       = out + (size_t)Bn * Vn;
  float* c1       = h1 + (size_t)Bn * Hn;
  float* attn     = c1 + (size_t)Bn * Hn;

  // Workspace layout
  float* ws     = (float*)d_ws;
  float* x_cat  = ws;                            // 64 x 3072
  float* gates  = x_cat + (size_t)Bn * XK;       // 64 x 4096
  float* logits = gates + (size_t)Bn * GN;       // 64 x 32000

  attn_ctx_kernel<<<Bn, 256, 0, stream>>>(input_batch, prev_h, enc, emb,
                                          attn_w, attn_b, x_cat, attn);
  gates_gemm_kernel<<<GN / 128, 256, 0, stream>>>(x_cat, W_ih, W_hh,
                                                  b_ih, b_hh, gates);
  lstm_kernel<<<Bn, 256, 0, stream>>>(gates, prev_c, lengths, h1, c1);
  logits_gemm_kernel<<<Vn / 128, 256, 0, stream>>>(h1, W_out, b_out, logits);
  logsoftmax_kernel<<<Bn, 512, 0, stream>>>(logits, logprobs);
}